// MemoryLayer_82566451298989
// MI455X (gfx1250) — compile-verified
//
#include <hip/hip_runtime.h>

// Problem constants (match reference)
#define BATCH_N   8192          // B*S
#define HIDDEN_K  1024
#define TOTAL_D   640           // NUM_TABLE * CODE_LEN
#define NUM_TAB   64
#define CODE_L    10
#define TAB_SZ    1024          // 2^CODE_LEN
#define OUT_D     1024

#define KC        64            // K-chunk staged in LDS per double-buffer slot

typedef float v2f __attribute__((ext_vector_type(2)));
typedef float v8f __attribute__((ext_vector_type(8)));
typedef int   v4i __attribute__((vector_size(4 * sizeof(int))));

#define GLOBAL_AS __attribute__((address_space(1)))
#define LDS_AS    __attribute__((address_space(3)))

#if __has_builtin(__builtin_amdgcn_global_load_async_to_lds_b128)
#define HAVE_ASYNC_LDS 1
#else
#define HAVE_ASYNC_LDS 0
#endif

__device__ __forceinline__ void wait_async_zero() {
#if HAVE_ASYNC_LDS
#if __has_builtin(__builtin_amdgcn_s_wait_asynccnt)
    __builtin_amdgcn_s_wait_asynccnt(0);
#else
    asm volatile("s_wait_asynccnt 0" ::: "memory");
#endif
#endif
}

// ---------------------------------------------------------------------------
// Kernel 1: z = x @ W + b via V_WMMA_F32_16X16X4_F32.
// Block = 128 threads = 4 waves; each wave owns one 16x16 output tile
// (64 rows x 16 cols per block). The 16-column B tile is shared by all
// 4 waves, so it is staged into LDS with async-to-LDS b128 copies,
// double-buffered over K-chunks of 64 and gated by s_wait_asynccnt+barrier.
// A stays as contiguous 8B global loads per lane (L2-friendly, reused 40x
// across column tiles).
// ---------------------------------------------------------------------------
__global__ __launch_bounds__(128) void gemm_z_wmma(
    const float* __restrict__ x,     // [N, HIDDEN]
    const float* __restrict__ W,     // [HIDDEN, TOTAL_D]
    const float* __restrict__ bvec,  // [TOTAL_D]
    float* __restrict__ z)           // [N, TOTAL_D]
{
    __shared__ float Btile[2][KC * 16];   // [k][n] row-major, 4 KB per buffer

    const int tid  = threadIdx.x;
    const int lane = tid & 31;
    const int wave = tid >> 5;
    const int half = lane >> 4;      // 0: K pair {0,1}, 1: K pair {2,3}
    const int l    = lane & 15;

    const int n0 = blockIdx.x * 16;                // column tile base
    const int m0 = (blockIdx.y * 4 + wave) * 16;   // row tile base

    const float* __restrict__ arow = x + (size_t)(m0 + l) * HIDDEN_K + 2 * half;

    // Cooperative stage of one KC x 16 B-chunk into LDS buffer `buf`.
    // 1024 floats = 256 float4; 128 threads x 2 b128 transfers.
    auto stage_chunk = [&](int k0, int buf) {
        float* dst = &Btile[buf][0];
#pragma unroll
        for (int j = 0; j < 2; ++j) {
            const int f  = tid + j * 128;      // float4 index in chunk
            const int k  = f >> 2;             // k-row within chunk
            const int n4 = (f & 3) << 2;       // column group
            const float* g = W + (size_t)(k0 + k) * TOTAL_D + n0 + n4;
            float* lp = dst + k * 16 + n4;
#if HAVE_ASYNC_LDS
            __builtin_amdgcn_global_load_async_to_lds_b128(
                (GLOBAL_AS v4i*)g, (LDS_AS v4i*)lp, 0, 0);
#else
            *(float4*)lp = *(const float4*)g;
#endif
        }
    };

    stage_chunk(0, 0);   // prefetch first chunk

    v8f acc = {};
    for (int c = 0; c < HIDDEN_K / KC; ++c) {
        wait_async_zero();
        __syncthreads();                       // chunk c resident in buf c&1
        if (c + 1 < HIDDEN_K / KC)
            stage_chunk((c + 1) * KC, (c + 1) & 1);

        const float* __restrict__ bb = &Btile[c & 1][0];
        const int kbase = c * KC;
#pragma unroll
        for (int kk = 0; kk < KC; kk += 4) {
            // A: 16x4 f32 tile; lane holds K = kbase+kk+2*half+{0,1} of row m0+l
            v2f a;
            a.x = arow[kbase + kk + 0];
            a.y = arow[kbase + kk + 1];
            // B from LDS: lane holds column n0+l, K = kk+2*half+{0,1}
            const int kb = kk + 2 * half;
            v2f b;
            b.x = bb[kb * 16 + l];
            b.y = bb[kb * 16 + 16 + l];
            acc = __builtin_amdgcn_wmma_f32_16x16x4_f32(
                    /*neg_a=*/false, a, /*neg_b=*/false, b,
                    /*c_mod=*/(short)0, acc, /*reuse_a=*/false, /*reuse_b=*/false);
        }
        __syncthreads();   // all waves done with buf c&1 before it is rewritten
    }

    // D layout: VGPR v -> (M = m0 + v + 8*half, N = n0 + l)
    const float bbias = bvec[n0 + l];
    float* __restrict__ zout = z + (size_t)(m0 + 8 * half) * TOTAL_D + (n0 + l);
#pragma unroll
    for (int v = 0; v < 8; ++v)
        zout[(size_t)v * TOTAL_D] = acc[v] + bbias;
}

// ---------------------------------------------------------------------------
// Kernel 2: per (row, table) -> bucket code (sign bits) + score (tanh product)
// ---------------------------------------------------------------------------
__global__ __launch_bounds__(256) void code_score_kernel(
    const float* __restrict__ z,     // [N, TOTAL_D]
    int*   __restrict__ code,        // [N * NUM_TAB]
    float* __restrict__ score)       // [N * NUM_TAB]
{
    const int idx = blockIdx.x * blockDim.x + threadIdx.x;   // n*64 + t
    if (idx >= BATCH_N * NUM_TAB) return;
    const int n = idx >> 6;
    const int t = idx & (NUM_TAB - 1);
    const float* __restrict__ zp = z + (size_t)n * TOTAL_D + t * CODE_L;

    float s = 1.0f;
    int   c = 0;
#pragma unroll
    for (int i = 0; i < CODE_L; ++i) {
        const float v  = zp[i];
        const float th = tanhf(fabsf(v));
        s *= (1.0f + th) * 0.5f;
        c |= (v > 0.0f) ? (1 << i) : 0;
    }
    code[idx]  = c;
    score[idx] = s;
}

// ---------------------------------------------------------------------------
// Kernel 3: out[n,:] = bias + sum_t score[n,t] * tables[t*1024 + code[n,t], :]
// One block per row; 256 threads x float4 => 1024 cols, b128 coalesced loads.
// 524288 gathers over 65536 distinct 4KB rows => ~8x reuse; default RT
// caching lets the 192MB L2 capture most of it.
// ---------------------------------------------------------------------------
__global__ __launch_bounds__(256) void gather_sum_kernel(
    const float* __restrict__ tables,  // [NUM_TAB*TAB_SZ, OUT_D]
    const float* __restrict__ bias,    // [OUT_D]
    const int*   __restrict__ code,    // [N * NUM_TAB]
    const float* __restrict__ score,   // [N * NUM_TAB]
    float* __restrict__ out)           // [N, OUT_D]
{
    __shared__ int   sc[NUM_TAB];
    __shared__ float ss[NUM_TAB];

    const int n   = blockIdx.x;
    const int tid = threadIdx.x;
    if (tid < NUM_TAB) {
        sc[tid] = code[(size_t)n * NUM_TAB + tid];
        ss[tid] = score[(size_t)n * NUM_TAB + tid];
    }
    __syncthreads();

    const int col = tid * 4;
    float4 acc = *(const float4*)(bias + col);

#pragma unroll 4
    for (int t = 0; t < NUM_TAB; ++t) {
        const size_t row = (size_t)(t << CODE_L) + (size_t)sc[t];   // t*1024 + code
        const float4 v = *(const float4*)(tables + row * OUT_D + col);
        const float  s = ss[t];
        acc.x += v.x * s;
        acc.y += v.y * s;
        acc.z += v.z * s;
        acc.w += v.w * s;
    }
    *(float4*)(out + (size_t)n * OUT_D + col) = acc;
}

// ---------------------------------------------------------------------------
// Launch
// ---------------------------------------------------------------------------
extern "C" void kernel_launch(void* const* d_in, const int* in_sizes, int n_in,
                              void* d_out, int out_size, void* d_ws, size_t ws_size,
                              hipStream_t stream) {
    const float* x      = (const float*)d_in[0];   // hidden_states [4,2048,1024]
    const float* W      = (const float*)d_in[1];   // W_proj [1024, 640]
    const float* b_proj = (const float*)d_in[2];   // [640]
    const float* tables = (const float*)d_in[3];   // [65536, 1024]
    const float* bias   = (const float*)d_in[4];   // [1024]
    float* out          = (float*)d_out;           // [8192, 1024]

    // Workspace layout: z (20 MB) | code (2 MB) | score (2 MB)
    const size_t Z_BYTES = (size_t)BATCH_N * TOTAL_D * sizeof(float);
    const size_t C_BYTES = (size_t)BATCH_N * NUM_TAB * sizeof(int);
    float* z     = (float*)d_ws;
    int*   code  = (int*)((char*)d_ws + Z_BYTES);
    float* score = (float*)((char*)d_ws + Z_BYTES + C_BYTES);

    // Phase 1: projection GEMM on the WMMA pipe with async-LDS staged B
    dim3 g1(TOTAL_D / 16, BATCH_N / 64);
    gemm_z_wmma<<<g1, 128, 0, stream>>>(x, W, b_proj, z);

    // Phase 2: codes + scores
    const int cs_total = BATCH_N * NUM_TAB;
    code_score_kernel<<<(cs_total + 255) / 256, 256, 0, stream>>>(z, code, score);

    // Phase 3: bandwidth-bound gather + weighted sum
    gather_sum_kernel<<<BATCH_N, 256, 0, stream>>>(tables, bias, code, score, out);
}